// UltraE_781684048760
// MI455X (gfx1250) — compile-verified
//
#include <hip/hip_runtime.h>
#include <math.h>

#define DIMV     512
#define KDIM     128
#define NBATCH   512
#define EPSF     1e-6f
#define MARGINF  2.0f

typedef __attribute__((ext_vector_type(2))) float v2f;
typedef __attribute__((ext_vector_type(8))) float v8f;

__global__ __launch_bounds__(256)
void ultrae_score_kernel(const float* __restrict__ emb,
                         const float* __restrict__ rel_boost,
                         const float* __restrict__ rel_left,
                         const float* __restrict__ rel_right,
                         const float* __restrict__ bias_head,
                         const float* __restrict__ bias_tail,
                         const int*   __restrict__ u_idx,
                         const int*   __restrict__ r_idx,
                         const int*   __restrict__ v_idx,
                         float*       __restrict__ out)
{
    __shared__ __attribute__((aligned(16))) float sh[DIMV];   // head staging
    __shared__ float s_red[8];
    __shared__ float s_scal[4];                               // a_h, ht0, ht1, nx
    __shared__ __attribute__((aligned(16))) float sS[8][16];  // per-wave dot results

    const int b    = blockIdx.x;
    const int tid  = threadIdx.x;
    const int lane = tid & 31;
    const int wave = tid >> 5;

    const int u = u_idx[b];
    const int r = r_idx[b];
    const float* headrow = emb + (size_t)u * DIMV;

    // ---------------- Phase 1: head transform ----------------
    // proj_tan0: element 0 forced to zero
    float h_a = (tid == 0) ? 0.0f : headrow[tid];
    float h_b = headrow[tid + 256];

    // a_h = -u1^2 + sum_{i>=2} u_i^2   (element tid and tid+256 per thread)
    float part = h_b * h_b + ((tid < 2) ? -(h_a * h_a) : (h_a * h_a));
    #pragma unroll
    for (int off = 16; off > 0; off >>= 1) part += __shfl_down(part, off);
    if (lane == 0) s_red[wave] = part;
    __syncthreads();
    if (tid == 0) {
        float a = 0.0f;
        #pragma unroll
        for (int w = 0; w < 8; ++w) a += s_red[w];
        s_scal[0] = a;
    }
    __syncthreads();

    // expmap0
    {
        float a_h = s_scal[0];
        float sp = sqrtf(fmaxf(a_h, EPSF));
        float st = sqrtf(fmaxf(-a_h, EPSF));
        float ch, sl;
        if (a_h > 0.0f) { ch = coshf(sp); sl = sinhf(sp) / sp; }
        else            { ch = cosf(st);  sl = sinf(st) / st; }
        sh[tid]       = (tid == 0) ? ch : sl * h_a;
        sh[tid + 256] = sl * h_b;
    }
    __syncthreads();

    // Givens rotation with rel_right: thread tid owns pair (2*tid, 2*tid+1)
    {
        const float* rr = rel_right + (size_t)r * DIMV;
        float g0 = rr[2 * tid], g1 = rr[2 * tid + 1];
        float n  = fmaxf(sqrtf(g0 * g0 + g1 * g1), 1e-15f);
        float c = g0 / n, s = g1 / n;
        float x0 = sh[2 * tid], x1 = sh[2 * tid + 1];
        sh[2 * tid]     = c * x0 - s * x1;
        sh[2 * tid + 1] = c * x1 + s * x0;
    }
    __syncthreads();

    // Sparse boost: dims {0,1} x {510,511}
    if (tid < 2) {
        float x  = rel_boost[(size_t)r * 2 + tid];
        float bb = fmaxf(x, 0.0f) + log1pf(expf(-fabsf(x)));  // softplus
        float Cc = sqrtf(1.0f + bb * bb);
        float xt = sh[tid], xb = sh[DIMV - 2 + tid];
        sh[tid]            = Cc * xt - bb * xb;
        sh[DIMV - 2 + tid] = Cc * xb - bb * xt;
    }
    __syncthreads();

    // Givens reflection with rel_left
    {
        const float* rl = rel_left + (size_t)r * DIMV;
        float g0 = rl[2 * tid], g1 = rl[2 * tid + 1];
        float n  = fmaxf(sqrtf(g0 * g0 + g1 * g1), 1e-15f);
        float c = g0 / n, s = g1 / n;
        float x0 = sh[2 * tid], x1 = sh[2 * tid + 1];
        sh[2 * tid]     = c * x0 + s * x1;
        sh[2 * tid + 1] = s * x0 - c * x1;
    }
    __syncthreads();

    if (tid == 0) {
        float ht0 = sh[0], ht1 = sh[1];
        s_scal[1] = ht0;
        s_scal[2] = ht1;
        s_scal[3] = sqrtf(ht0 * ht0 + ht1 * ht1);
        sh[0] = 0.0f;  // zero time dims so WMMA dot is spatial-only
        sh[1] = 0.0f;
    }
    __syncthreads();

    // ---------------- Phase 2: WMMA dot vs 16 tails per wave ----------------
    const int toff = wave * 16 + (lane & 15);
    const int vid  = v_idx[(size_t)b * KDIM + toff];
    const int off  = (lane >= 16) ? 2 : 0;
    const float* trow = emb + (size_t)vid * DIMV + off;
    const float* hrow = sh + off;

    float q  = 0.0f;   // spatial sum-of-squares of raw tail
    float t1 = 0.0f;   // raw tail element 1
    v8f acc = {};
    #pragma unroll 4
    for (int c = 0; c < DIMV / 4; ++c) {
        v2f av = *(const v2f*)(trow + 4 * c);   // A: 16 tails x 4 chunk
        v2f bv = *(const v2f*)(hrow + 4 * c);   // B: head chunk, all 16 cols equal
        bool skip = (c == 0) && (lane < 16);    // exclude elements 0,1 from q
        float sq = av.x * av.x + av.y * av.y;
        q += skip ? 0.0f : sq;
        if (c == 0 && lane < 16) t1 = av.y;
        // D = A*B + C  (full fp32 matrix op)
        acc = __builtin_amdgcn_wmma_f32_16x16x4_f32(false, av, false, bv,
                                                    (short)0, acc, false, false);
    }

    // C layout: lane 0 VGPR r = S_row r ; lane 16 VGPR r = S_row r+8
    if (lane == 0 || lane == 16) {
        const int base = (lane == 0) ? 0 : 8;
        #pragma unroll
        for (int m = 0; m < 8; ++m) sS[wave][base + m] = acc[m];
    }
    q += __shfl_xor(q, 16);   // combine elem {0,1 mod 4} and {2,3 mod 4} halves

    if (lane < 16) {
        float S   = sS[wave][lane];
        float a_t = q - t1 * t1;                 // -t1^2 + sum_{i>=2} t_i^2
        float sp  = sqrtf(fmaxf(a_t, EPSF));
        float st  = sqrtf(fmaxf(-a_t, EPSF));
        float c_t, s_t;
        if (a_t > 0.0f) { c_t = coshf(sp); s_t = sinhf(sp) / sp; }
        else            { c_t = cosf(st);  s_t = sinf(st) / st; }
        float th0 = c_t;
        float th1 = s_t * t1;
        float ny  = sqrtf(th0 * th0 + th1 * th1);
        float ht0 = s_scal[1], ht1 = s_scal[2], nx = s_scal[3];
        float cosang = (ht0 * th0 + ht1 * th1) / (nx * ny);
        cosang = fminf(fmaxf(cosang, -1.0f + EPSF), 1.0f - EPSF);
        float theta = acosf(cosang);
        float inner = fmaxf(nx * ny - s_t * S, 1.0f + EPSF);
        float dhyp  = acoshf(inner);
        float base  = fminf(nx, ny) * theta + dhyp;  // min-norm angular arc + hyperbolic dist
        float dist  = base * base;
        out[(size_t)b * KDIM + toff] =
            MARGINF - dist + bias_head[u] + bias_tail[vid];
    }
}

extern "C" void kernel_launch(void* const* d_in, const int* in_sizes, int n_in,
                              void* d_out, int out_size, void* d_ws, size_t ws_size,
                              hipStream_t stream) {
    const float* emb = (const float*)d_in[0];
    const float* rb  = (const float*)d_in[1];
    const float* rl  = (const float*)d_in[2];
    const float* rr  = (const float*)d_in[3];
    const float* bh  = (const float*)d_in[4];
    const float* bt  = (const float*)d_in[5];
    const int*   ui  = (const int*)d_in[6];
    const int*   ri  = (const int*)d_in[7];
    const int*   vi  = (const int*)d_in[8];
    float* outp = (float*)d_out;
    ultrae_score_kernel<<<NBATCH, 256, 0, stream>>>(emb, rb, rl, rr, bh, bt,
                                                    ui, ri, vi, outp);
}